// AbstractMultilayerClustering_47373489275294
// MI455X (gfx1250) — compile-verified
//
#include <hip/hip_runtime.h>

// Two-level nearest-centroid for MI455X (gfx1250, wave32).
// Stage 1: D[center][point] = ||c||^2 - 2 c.x via V_WMMA_F32_16X16X4_F32,
//   centers (+ norms through an augmented K-step vs constant-1) resident in
//   VGPRs; argmin = in-register compare/select scan + one xor-16 shuffle.
// Stage 2: 8 inner centers per point, k-vectorized out of LDS (b128 chunks,
//   distances materialize as two f4 vectors -> no horizontal sums).
// Memory-bound target: ~132 MB @ 23.3 TB/s.

typedef float v2f __attribute__((ext_vector_type(2)));
typedef float v8f __attribute__((ext_vector_type(8)));
typedef float f4  __attribute__((ext_vector_type(4)));

#define NPTS     1000000
#define NCHUNKS  ((NPTS + 255) / 256)   // 3907 chunks of 256 points
#define NBLOCKS  768                    // persistent-ish, grid-stride

__global__ __launch_bounds__(256) void mlc_kernel(
    const float* __restrict__ x,    // [N][32] : x1 = [:,0:16], x2 = [:,16:32]
    const float* __restrict__ co,   // [64][16]
    const float* __restrict__ ci,   // [64][8][16]
    float*       __restrict__ out)  // [N] indices as float
{
    // k-vectorized inner centers: s_ciT[(e*2+g)*64 + o][q] = -2*ci[o][g*4+q][e]
    // (e = feature 0..15, g = k-group 0..1, q = k within group).
    // b128 reads index (e*2+g)*64 + o -> bank quad = o%16 (<=2-way conflicts).
    __shared__ f4    s_ciT[16 * 2 * 64];
    __shared__ float s_cinorm[512];     // [o*8 + k]
    __shared__ float s_onorm[64];

    const int tid  = threadIdx.x;
    const int lane = tid & 31;
    const int wave = tid >> 5;

    // ---- one-time staging ----
    for (int i = tid; i < 512; i += 256) {
        const int o = i >> 3, k = i & 7;
        const int g = k >> 2, q = k & 3;
        const float* src = ci + i * 16;
        float nrm = 0.f;
        #pragma unroll
        for (int e = 0; e < 16; ++e) {
            float v = src[e];
            nrm += v * v;
            s_ciT[(e * 2 + g) * 64 + o][q] = -2.f * v;
        }
        s_cinorm[i] = nrm;
    }
    if (tid < 64) {
        const float* src = co + tid * 16;
        float nrm = 0.f;
        #pragma unroll
        for (int j = 0; j < 16; ++j) nrm += src[j] * src[j];
        s_onorm[tid] = nrm;
    }
    __syncthreads();

    const int n  = lane & 15;   // A: center-in-tile row; B/D: point column
    const int kh = lane >> 4;   // K-half (lanes 16-31 hold K offset +2)

    // A = -2 * centers_outer (M=center, K=feature), VGPR-resident.
    v2f Am[4][4];
    #pragma unroll
    for (int t = 0; t < 4; ++t)
        #pragma unroll
        for (int j = 0; j < 4; ++j) {
            const float* p = co + (t * 16 + n) * 16 + 4 * j + 2 * kh;
            Am[t][j][0] = -2.f * p[0];
            Am[t][j][1] = -2.f * p[1];
        }
    // Augmented K-step: A[m][16] = ||c_m||^2 vs B[16][n] = 1 -> seeds C with norms.
    v2f A4[4], Bc;
    #pragma unroll
    for (int t = 0; t < 4; ++t) { A4[t][0] = kh ? 0.f : s_onorm[t * 16 + n]; A4[t][1] = 0.f; }
    Bc[0] = kh ? 0.f : 1.f;  Bc[1] = 0.f;

    const v8f Zero = {};   // inline-0 C operand

    for (int chunk = blockIdx.x; chunk < NCHUNKS; chunk += (int)gridDim.x) {
        const int base = chunk * 256 + wave * 32;   // 32 points per wave

        int otile0 = 0, otile1 = 0;
        #pragma unroll
        for (int half = 0; half < 2; ++half) {
            int prow = base + half * 16 + n;
            if (prow > NPTS - 1) prow = NPTS - 1;            // tail clamp
            // B = x1^T tile (K=feature, N=point)
            v2f Bp[4];
            #pragma unroll
            for (int j = 0; j < 4; ++j) {
                const float* p = x + (size_t)prow * 32 + 4 * j + 2 * kh;
                Bp[j][0] = p[0];
                Bp[j][1] = p[1];
            }
            v8f D[4];
            #pragma unroll
            for (int t = 0; t < 4; ++t) {
                v8f c = __builtin_amdgcn_wmma_f32_16x16x4_f32(
                            false, A4[t], false, Bc, (short)0, Zero, false, false);
                #pragma unroll
                for (int j = 0; j < 4; ++j)
                    c = __builtin_amdgcn_wmma_f32_16x16x4_f32(
                            false, Am[t][j], false, Bp[j], (short)0, c, false, false);
                D[t] = c;
            }
            // Lane holds scores of centers {t*16 + r + 8*kh} for point (base+half*16+n).
            // Compare/select scan in ascending center order (first-min tie-break);
            // plain ternaries -> cmp + dual-cndmask, no IEEE canonicalize ops.
            float bv = D[0][0];
            int   bi = 0;
            #pragma unroll
            for (int t = 0; t < 4; ++t)
                #pragma unroll
                for (int r = 0; r < 8; ++r) {
                    if (t == 0 && r == 0) continue;
                    float dv = D[t][r];
                    bool  lt = dv < bv;
                    bv = lt ? dv : bv;
                    bi = lt ? (t * 16 + r) : bi;
                }
            bi += kh * 8;                                    // full center index
            // Combine the two K-halves (lanes p and p+16 hold point p's halves).
            float ov = __shfl_xor(bv, 16, 32);
            int   oi = __shfl_xor(bi, 16, 32);
            bool take = (ov < bv) || (ov == bv && oi < bi);  // smaller-index tie-break
            bi = take ? oi : bi;
            if (half == 0) otile0 = bi; else otile1 = bi;
        }

        // ---- stage 2: one lane per point, k-vectorized ----
        const int o  = (lane < 16) ? otile0 : otile1;
        const int p  = base + lane;
        const int pc = (p > NPTS - 1) ? (NPTS - 1) : p;

        f4 xv[4];
        {
            const f4* src = (const f4*)(x + (size_t)pc * 32 + 16);
            #pragma unroll
            for (int j = 0; j < 4; ++j) xv[j] = __builtin_nontemporal_load(src + j);
        }

        f4 acc0 = {}, acc1 = {};
        #pragma unroll
        for (int e = 0; e < 16; ++e) {
            float xe = xv[e >> 2][e & 3];
            f4 c0 = s_ciT[(e * 2 + 0) * 64 + o];
            f4 c1 = s_ciT[(e * 2 + 1) * 64 + o];
            acc0 += c0 * xe;    // v_pk_fma_f32 with scalar broadcast
            acc1 += c1 * xe;
        }
        const f4* cnp = (const f4*)&s_cinorm[o * 8];
        f4 d0 = cnp[0] + acc0;   // distances for k = 0..3
        f4 d1 = cnp[1] + acc1;   // distances for k = 4..7

        float best = d0[0];
        int   bk   = 0;
        #pragma unroll
        for (int k = 1; k < 8; ++k) {
            float d  = (k < 4) ? d0[k] : d1[k - 4];
            bool  lt = d < best;
            best = lt ? d : best;
            bk   = lt ? k : bk;                              // first-min tie-break
        }
        if (p < NPTS) out[p] = (float)(o * 8 + bk);
    }
}

extern "C" void kernel_launch(void* const* d_in, const int* in_sizes, int n_in,
                              void* d_out, int out_size, void* d_ws, size_t ws_size,
                              hipStream_t stream) {
    const float* x  = (const float*)d_in[0];   // [1000000, 32]
    const float* co = (const float*)d_in[1];   // [64, 16]
    const float* ci = (const float*)d_in[2];   // [64, 8, 16]
    float* out = (float*)d_out;                // [1000000] indices as float

    mlc_kernel<<<NBLOCKS, 256, 0, stream>>>(x, co, ci, out);
}